// gin_conv_14250701488895
// MI455X (gfx1250) — compile-verified
//
#include <hip/hip_runtime.h>

typedef float v2f __attribute__((ext_vector_type(2)));
typedef float v8f __attribute__((ext_vector_type(8)));

#define NNODES 50000
#define NEDGES 800000
#define FEAT   128
#define BN_EPS 1e-5f

// 50000 rows = 625 blocks * 5 row-tiles * 16 rows
#define ROW_TILES 5
#define ROWS_PER_BLOCK (ROW_TILES * 16)   // 80
#define LDS_STRIDE 132                    // 128 + 4 pad: lanes hit distinct banks

// ---------------------------------------------------------------------------
// Kernel 1: y = (1 + eps) * x   (residual pre-seeded so scatter-add lands on it)
// ---------------------------------------------------------------------------
__global__ void k_init_y(const float* __restrict__ x,
                         const float* __restrict__ eps,
                         float* __restrict__ y) {
    size_t i = (size_t)blockIdx.x * blockDim.x + threadIdx.x;
    float s = 1.0f + eps[0];
    const float4* xv = (const float4*)x;
    float4* yv = (float4*)y;
    float4 v = xv[i];
    v.x *= s; v.y *= s; v.z *= s; v.w *= s;
    yv[i] = v;
}

// ---------------------------------------------------------------------------
// Kernel 2: zero the 256-float stats buffer (sum[128], sumsq[128])
// ---------------------------------------------------------------------------
__global__ void k_zero_stats(float* __restrict__ stats) {
    stats[threadIdx.x] = 0.0f;
}

// ---------------------------------------------------------------------------
// Kernel 3: SpMM scatter: y[dst] += w * x[src]
// one wave (32 lanes) per edge, each lane owns 4 consecutive features (float4)
// ---------------------------------------------------------------------------
__global__ void k_spmm_scatter(const float* __restrict__ x,
                               const float* __restrict__ ew,
                               const int* __restrict__ esrc,
                               const int* __restrict__ edst,
                               float* __restrict__ y) {
    int e = blockIdx.x * 8 + (threadIdx.x >> 5);
    int lane = threadIdx.x & 31;
    int s = esrc[e];
    int d = edst[e];
    float w = ew[e];
    const float4* xs = (const float4*)(x + (size_t)s * FEAT);
    float4 v = xs[lane];
    float* yd = y + (size_t)d * FEAT + lane * 4;
    atomicAdd(yd + 0, v.x * w);
    atomicAdd(yd + 1, v.y * w);
    atomicAdd(yd + 2, v.z * w);
    atomicAdd(yd + 3, v.w * w);
}

// ---------------------------------------------------------------------------
// Kernel 4: C = act(A @ W + b), optional per-column sum / sumsq accumulation.
// A: [NNODES,128] f32, W: [128,128] row-major, C: [NNODES,128].
//
// Block = 80 rows (5 x 16-row tiles) staged in LDS with b128 loads.
// 8 waves each own one 16-column tile; the wave's B panel (128x16 of W) is
// loaded ONCE into 32 v2f registers and reused across all 5 row tiles.
// Inner loop per tile: 32 x V_WMMA_F32_16X16X4_F32, A fed by ds_load_2addr_b64.
//
// f32 16x16x4 VGPR layouts (ISA 7.12.2):
//   A 16x4 : lane<16 -> M=lane, v0:K=0 v1:K=1 ; lane>=16 -> v0:K=2 v1:K=3
//   B 4x16 : lane<16 -> N=lane, v0:K=0 v1:K=1 ; lane>=16 -> v0:K=2 v1:K=3
//   C 16x16: c[r] = C[M = r + 8*(lane/16), N = lane%16]
// ---------------------------------------------------------------------------
__global__ void k_gemm_bias(const float* __restrict__ A,
                            const float* __restrict__ W,
                            const float* __restrict__ bias,
                            float* __restrict__ out,
                            float* __restrict__ stats,   // null => no BN stats
                            int do_relu) {
    __shared__ __align__(16) float As[ROWS_PER_BLOCK * LDS_STRIDE];

    const int wave = threadIdx.x >> 5;
    const int lane = threadIdx.x & 31;
    const int half = lane >> 4;
    const int lid  = lane & 15;
    const int m0   = blockIdx.x * ROWS_PER_BLOCK;
    const int n0   = wave * 16;

    // stage the 80x128 A panel into LDS with wide b128 loads
    const float4* Ag = (const float4*)(A + (size_t)m0 * FEAT);
    for (int i4 = threadIdx.x; i4 < ROWS_PER_BLOCK * FEAT / 4; i4 += 256) {
        int r = i4 >> 5;              // /32 float4 per row
        int c = (i4 & 31) << 2;
        *(float4*)&As[r * LDS_STRIDE + c] = Ag[i4];
    }

    // hoist the wave's B panel (W columns n0..n0+15) into registers
    const float* Wc = W + n0 + lid;
    v2f breg[32];
#pragma unroll
    for (int kk = 0; kk < 32; ++kk) {
        int k = 4 * kk + 2 * half;
        breg[kk].x = Wc[(size_t)k * FEAT];
        breg[kk].y = Wc[(size_t)(k + 1) * FEAT];
    }
    const float bv = bias[n0 + lid];

    __syncthreads();

    float s1 = 0.0f, s2 = 0.0f;
#pragma unroll
    for (int t = 0; t < ROW_TILES; ++t) {
        const float* Ar = As + (t * 16 + lid) * LDS_STRIDE + 2 * half;
        v8f c = {};
#pragma unroll
        for (int kk = 0; kk < 32; ++kk) {
            v2f a;
            a.x = Ar[4 * kk];
            a.y = Ar[4 * kk + 1];
            c = __builtin_amdgcn_wmma_f32_16x16x4_f32(
                    /*neg_a=*/false, a, /*neg_b=*/false, breg[kk],
                    /*c_mod=*/(short)0, c, /*reuse_a=*/false, /*reuse_b=*/false);
        }
#pragma unroll
        for (int r = 0; r < 8; ++r) {
            float v = c[r] + bv;
            if (do_relu) v = fmaxf(v, 0.0f);
            int m = m0 + t * 16 + r + 8 * half;
            out[(size_t)m * FEAT + n0 + lid] = v;
            s1 += v;
            s2 += v * v;
        }
    }

    if (stats) {
        // fold the two 16-lane halves (same column, different rows)
        s1 += __shfl_xor(s1, 16, 32);
        s2 += __shfl_xor(s2, 16, 32);
        if (half == 0) {
            atomicAdd(&stats[n0 + lid], s1);           // column sums
            atomicAdd(&stats[FEAT + n0 + lid], s2);    // column sums of squares
        }
    }
}

// ---------------------------------------------------------------------------
// Kernel 5: fold sums into affine scale/shift (128 threads, 1 block)
// ---------------------------------------------------------------------------
__global__ void k_bn_finalize(const float* __restrict__ stats,
                              const float* __restrict__ gamma,
                              const float* __restrict__ beta,
                              float* __restrict__ scale_shift) {
    int f = threadIdx.x;
    float inv_n = 1.0f / (float)NNODES;
    float mean = stats[f] * inv_n;
    float var  = stats[FEAT + f] * inv_n - mean * mean;
    float sc   = gamma[f] * rsqrtf(var + BN_EPS);
    scale_shift[f]        = sc;
    scale_shift[FEAT + f] = beta[f] - mean * sc;
}

// ---------------------------------------------------------------------------
// Kernel 6: in-place normalize d_out: v = v*scale[col] + shift[col]
// ---------------------------------------------------------------------------
__global__ void k_bn_apply(float* __restrict__ out,
                           const float* __restrict__ scale_shift) {
    size_t i = (size_t)blockIdx.x * blockDim.x + threadIdx.x;
    int c0 = ((int)(i & 31)) * 4;
    float4* ov = (float4*)out;
    float4 v = ov[i];
    const float* sc = scale_shift;
    const float* sh = scale_shift + FEAT;
    v.x = v.x * sc[c0 + 0] + sh[c0 + 0];
    v.y = v.y * sc[c0 + 1] + sh[c0 + 1];
    v.z = v.z * sc[c0 + 2] + sh[c0 + 2];
    v.w = v.w * sc[c0 + 3] + sh[c0 + 3];
    ov[i] = v;
}

// ---------------------------------------------------------------------------
extern "C" void kernel_launch(void* const* d_in, const int* in_sizes, int n_in,
                              void* d_out, int out_size, void* d_ws, size_t ws_size,
                              hipStream_t stream) {
    const float* x     = (const float*)d_in[0];
    const float* ew    = (const float*)d_in[1];
    const float* W1    = (const float*)d_in[2];
    const float* b1    = (const float*)d_in[3];
    const float* W2    = (const float*)d_in[4];
    const float* b2    = (const float*)d_in[5];
    const float* eps   = (const float*)d_in[6];
    const float* gamma = (const float*)d_in[7];
    const float* beta  = (const float*)d_in[8];
    const int* esrc    = (const int*)d_in[9];
    const int* edst    = (const int*)d_in[10];
    float* out = (float*)d_out;

    const size_t mat_bytes = (size_t)NNODES * FEAT * sizeof(float); // 25.6 MB
    float* y     = (float*)d_ws;                                     // [N,128]
    float* t     = (float*)((char*)d_ws + mat_bytes);                // [N,128]
    float* stats = (float*)((char*)d_ws + 2 * mat_bytes);            // 256 f32
    float* ss    = stats + 256;                                      // scale/shift

    const int elems4 = NNODES * FEAT / 4;                            // 1.6M

    k_init_y<<<elems4 / 256, 256, 0, stream>>>(x, eps, y);
    k_zero_stats<<<1, 256, 0, stream>>>(stats);
    k_spmm_scatter<<<NEDGES / 8, 256, 0, stream>>>(x, ew, esrc, edst, y);
    k_gemm_bias<<<NNODES / ROWS_PER_BLOCK, 256, 0, stream>>>(y, W1, b1, t, nullptr, 1);
    k_gemm_bias<<<NNODES / ROWS_PER_BLOCK, 256, 0, stream>>>(t, W2, b2, out, stats, 0);
    k_bn_finalize<<<1, FEAT, 0, stream>>>(stats, gamma, beta, ss);
    k_bn_apply<<<elems4 / 256, 256, 0, stream>>>(out, ss);
}